// SSM_44667659878983
// MI455X (gfx1250) — compile-verified
//
#include <hip/hip_runtime.h>
#include <hip/hip_bf16.h>
#include <math.h>

// ---------------------------------------------------------------------------
// SSM: Bu = x @ B^T ; state_t = A state_{t-1} + Bu_t ; y = states @ C^T ;
//      y = LayerNorm(GELU(y))
// Memory-bound (~140MB @ 23.3TB/s ~ 6us). Full-f32 WMMA (v_wmma_f32_16x16x4_f32)
// for both GEMMs; chunked linear-recurrence scan for parallelism.
// ---------------------------------------------------------------------------

typedef __attribute__((ext_vector_type(2))) float v2f;
typedef __attribute__((ext_vector_type(8))) float v8f;

#define D_MODEL 512
#define D_STATE 64
#define SEQ_L   4096
#define BATCH   8
#define M_TOT   (BATCH * SEQ_L)      // 32768 rows
#define TC      128                  // scan chunk length
#define NCHUNK  (SEQ_L / TC)         // 32 chunks

static __device__ __forceinline__ v8f wmma_f32(v2f a, v2f b, v8f c) {
    // D = A(16x4,f32) x B(4x16,f32) + C(16x16,f32)
    return __builtin_amdgcn_wmma_f32_16x16x4_f32(
        /*neg_a=*/false, a, /*neg_b=*/false, b,
        /*c_mod=*/(short)0, c, /*reuse_a=*/false, /*reuse_b=*/false);
}

// ---------------------------------------------------------------------------
// K1: Bu[m][n] = sum_d x[m][d] * B[n][d]   (M=32768, K=512, N=64)
// One wave -> 16x64 output strip (4 accumulators), K-loop of 128 x K=4 steps.
// ---------------------------------------------------------------------------
__global__ void ssm_k1_bu(const float* __restrict__ x,
                          const float* __restrict__ Bm,
                          float* __restrict__ Bu) {
    const int wave = (blockIdx.x * blockDim.x + threadIdx.x) >> 5;
    const int lane = threadIdx.x & 31;
    const int half = lane >> 4;       // 0: lanes 0-15, 1: lanes 16-31
    const int lr   = lane & 15;
    const int m0   = wave * 16;

    const float* __restrict__ xrow = x + (size_t)(m0 + lr) * D_MODEL;

    v8f acc[4];
    acc[0] = v8f{}; acc[1] = v8f{}; acc[2] = v8f{}; acc[3] = v8f{};

    for (int k0 = 0; k0 < D_MODEL; k0 += 4) {
        const int ka = k0 + 2 * half;
        v2f a;
        a[0] = xrow[ka];
        a[1] = xrow[ka + 1];
#pragma unroll
        for (int t = 0; t < 4; ++t) {
            const int col = t * 16 + lr;        // N column, 0..63
            v2f b;
            b[0] = Bm[col * D_MODEL + ka];
            b[1] = Bm[col * D_MODEL + ka + 1];
            acc[t] = wmma_f32(a, b, acc[t]);
        }
    }
    // D layout: VGPR i -> row (i + 8*half), col = n0 + lr
#pragma unroll
    for (int t = 0; t < 4; ++t) {
#pragma unroll
        for (int i = 0; i < 8; ++i) {
            Bu[(size_t)(m0 + i + 8 * half) * D_STATE + t * 16 + lr] = acc[t][i];
        }
    }
}

// ---------------------------------------------------------------------------
// K2: per-(batch,chunk) local scan with zero init, in-place over Bu.
//     loc_t = A loc_{t-1} + bu_t ; store chunk-end state to carry_end.
// 64 threads (2 waves): thread n owns state[n]; A staged in LDS.
// ---------------------------------------------------------------------------
__global__ void ssm_k2_localscan(const float* __restrict__ A,
                                 float* __restrict__ Bu,
                                 float* __restrict__ carry_end) {
    __shared__ float As[D_STATE * D_STATE];
    __shared__ float st[D_STATE];
    const int n = threadIdx.x;
    const int b = blockIdx.x >> 5;
    const int c = blockIdx.x & (NCHUNK - 1);

    for (int i = n; i < D_STATE * D_STATE; i += D_STATE) As[i] = A[i];
    st[n] = 0.0f;
    __syncthreads();

    float* __restrict__ bu = Bu + ((size_t)b * SEQ_L + (size_t)c * TC) * D_STATE;
    const float* __restrict__ ar = As + n * D_STATE;

    for (int j = 0; j < TC; ++j) {
        float acc = bu[j * D_STATE + n];
#pragma unroll
        for (int k = 0; k < D_STATE; ++k) acc = fmaf(ar[k], st[k], acc);
        __syncthreads();          // everyone done reading old state
        st[n] = acc;
        bu[j * D_STATE + n] = acc;
        __syncthreads();          // new state visible
    }
    carry_end[(size_t)(b * NCHUNK + c) * D_STATE + n] = st[n];
}

// ---------------------------------------------------------------------------
// K3: Apow = A^TC by repeated squaring (TC=128 -> 7 squarings). One block.
// ---------------------------------------------------------------------------
__global__ void ssm_k3_apow(const float* __restrict__ A,
                            float* __restrict__ Apow) {
    __shared__ float cur[D_STATE * D_STATE];
    __shared__ float nxt[D_STATE * D_STATE];
    const int tid = threadIdx.x; // 256 threads
    for (int i = tid; i < D_STATE * D_STATE; i += 256) cur[i] = A[i];
    __syncthreads();
    for (int it = 0; it < 7; ++it) {     // A^(2^7) = A^128
        for (int e = tid; e < D_STATE * D_STATE; e += 256) {
            const int i = e >> 6, j = e & 63;
            float acc = 0.0f;
#pragma unroll
            for (int k = 0; k < D_STATE; ++k)
                acc = fmaf(cur[i * D_STATE + k], cur[k * D_STATE + j], acc);
            nxt[e] = acc;
        }
        __syncthreads();
        for (int e = tid; e < D_STATE * D_STATE; e += 256) cur[e] = nxt[e];
        __syncthreads();
    }
    for (int i = tid; i < D_STATE * D_STATE; i += 256) Apow[i] = cur[i];
}

// ---------------------------------------------------------------------------
// K4: sequential carry across chunks: s_{c+1} = Apow s_c + carry_end[c].
// Records s_in[b][c] = incoming state of chunk c. One block, 512 threads.
// ---------------------------------------------------------------------------
__global__ void ssm_k4_carry(const float* __restrict__ Apow,
                             const float* __restrict__ carry_end,
                             float* __restrict__ s_in) {
    __shared__ float Ap[D_STATE * D_STATE];
    __shared__ float s[BATCH * D_STATE];
    const int tid = threadIdx.x;            // 0..511
    const int b = tid >> 6, n = tid & 63;
    for (int i = tid; i < D_STATE * D_STATE; i += 512) Ap[i] = Apow[i];
    s[tid] = 0.0f;
    __syncthreads();
    for (int c = 0; c < NCHUNK; ++c) {
        s_in[(size_t)(b * NCHUNK + c) * D_STATE + n] = s[tid];
        float acc = carry_end[(size_t)(b * NCHUNK + c) * D_STATE + n];
#pragma unroll
        for (int k = 0; k < D_STATE; ++k)
            acc = fmaf(Ap[n * D_STATE + k], s[b * D_STATE + k], acc);
        __syncthreads();
        s[tid] = acc;
        __syncthreads();
    }
}

// ---------------------------------------------------------------------------
// K5: fix-up: states[b][c*TC+j] += A^{j+1} s_in[b][c] via incremental matvec.
// ---------------------------------------------------------------------------
__global__ void ssm_k5_fixup(const float* __restrict__ A,
                             const float* __restrict__ s_in,
                             float* __restrict__ states) {
    __shared__ float As[D_STATE * D_STATE];
    __shared__ float v[D_STATE];
    const int n = threadIdx.x;
    const int b = blockIdx.x >> 5;
    const int c = blockIdx.x & (NCHUNK - 1);
    for (int i = n; i < D_STATE * D_STATE; i += D_STATE) As[i] = A[i];
    v[n] = s_in[(size_t)(b * NCHUNK + c) * D_STATE + n];
    __syncthreads();

    float* __restrict__ st = states + ((size_t)b * SEQ_L + (size_t)c * TC) * D_STATE;
    const float* __restrict__ ar = As + n * D_STATE;
    for (int j = 0; j < TC; ++j) {
        float acc = 0.0f;
#pragma unroll
        for (int k = 0; k < D_STATE; ++k) acc = fmaf(ar[k], v[k], acc);
        st[j * D_STATE + n] += acc;
        __syncthreads();
        v[n] = acc;
        __syncthreads();
    }
}

// ---------------------------------------------------------------------------
// K6: y = states @ C^T  (M=32768, K=64, N=512), fused GELU(exact) + LayerNorm.
// Block = 256 threads (8 waves), block owns a 16x512 row strip.
// Wave w computes cols [64w, 64w+64); tile staged in LDS; LN via shfl reduce.
// ---------------------------------------------------------------------------
__global__ void ssm_k6_out(const float* __restrict__ S,
                           const float* __restrict__ Cm,
                           const float* __restrict__ gamma,
                           const float* __restrict__ beta,
                           float* __restrict__ out) {
    __shared__ float tile[16 * D_MODEL]; // 32 KB
    const int wave = threadIdx.x >> 5;
    const int lane = threadIdx.x & 31;
    const int half = lane >> 4;
    const int lr   = lane & 15;
    const int m0   = blockIdx.x * 16;
    const int d0   = wave * 64;

    const float* __restrict__ srow = S + (size_t)(m0 + lr) * D_STATE;

    v8f acc[4];
    acc[0] = v8f{}; acc[1] = v8f{}; acc[2] = v8f{}; acc[3] = v8f{};

    for (int k0 = 0; k0 < D_STATE; k0 += 4) {
        const int ka = k0 + 2 * half;
        v2f a;
        a[0] = srow[ka];
        a[1] = srow[ka + 1];
#pragma unroll
        for (int t = 0; t < 4; ++t) {
            const int col = d0 + t * 16 + lr;     // output feature dim
            v2f b;
            b[0] = Cm[col * D_STATE + ka];
            b[1] = Cm[col * D_STATE + ka + 1];
            acc[t] = wmma_f32(a, b, acc[t]);
        }
    }

    // exact GELU, stage into LDS with D-matrix layout
#pragma unroll
    for (int t = 0; t < 4; ++t) {
#pragma unroll
        for (int i = 0; i < 8; ++i) {
            float vv = acc[t][i];
            vv = 0.5f * vv * (1.0f + erff(vv * 0.70710678118654752f));
            tile[(i + 8 * half) * D_MODEL + d0 + t * 16 + lr] = vv;
        }
    }
    __syncthreads();

    // LayerNorm: 8 waves, wave handles rows {wave, wave+8}
    for (int r = wave; r < 16; r += 8) {
        float s1 = 0.0f, s2 = 0.0f;
#pragma unroll
        for (int i = 0; i < 16; ++i) {
            const float vv = tile[r * D_MODEL + lane + i * 32];
            s1 += vv;
            s2 += vv * vv;
        }
#pragma unroll
        for (int off = 16; off >= 1; off >>= 1) {
            s1 += __shfl_xor(s1, off, 32);
            s2 += __shfl_xor(s2, off, 32);
        }
        const float mean = s1 * (1.0f / D_MODEL);
        const float var  = s2 * (1.0f / D_MODEL) - mean * mean;
        const float inv  = rsqrtf(var + 1e-5f);
        float* __restrict__ orow = out + (size_t)(m0 + r) * D_MODEL;
#pragma unroll
        for (int i = 0; i < 16; ++i) {
            const int col = lane + i * 32;
            const float vv = tile[r * D_MODEL + col];
            orow[col] = (vv - mean) * inv * gamma[col] + beta[col];
        }
    }
}

// ---------------------------------------------------------------------------
extern "C" void kernel_launch(void* const* d_in, const int* in_sizes, int n_in,
                              void* d_out, int out_size, void* d_ws, size_t ws_size,
                              hipStream_t stream) {
    const float* x     = (const float*)d_in[0]; // [8,4096,512]
    const float* A     = (const float*)d_in[1]; // [64,64]
    const float* Bm    = (const float*)d_in[2]; // [64,512]
    const float* Cm    = (const float*)d_in[3]; // [512,64]
    const float* gamma = (const float*)d_in[4]; // [512]
    const float* beta  = (const float*)d_in[5]; // [512]
    float* out = (float*)d_out;                 // [8,4096,512]

    // workspace layout (floats): Bu/states | carry_end | s_in | Apow
    float* Bu        = (float*)d_ws;                        // 2,097,152
    float* carry_end = Bu + (size_t)M_TOT * D_STATE;        // 16,384
    float* s_in      = carry_end + BATCH * NCHUNK * D_STATE;// 16,384
    float* Apow      = s_in + BATCH * NCHUNK * D_STATE;     // 4,096

    // K1: Bu = x @ B^T        (2048 m-tiles, 8 waves/block)
    ssm_k1_bu<<<dim3(M_TOT / 16 / 8), dim3(256), 0, stream>>>(x, Bm, Bu);
    // K3: A^128 (independent of K1/K2)
    ssm_k3_apow<<<dim3(1), dim3(256), 0, stream>>>(A, Apow);
    // K2: local scans (in-place over Bu)
    ssm_k2_localscan<<<dim3(BATCH * NCHUNK), dim3(D_STATE), 0, stream>>>(A, Bu, carry_end);
    // K4: sequential chunk carries
    ssm_k4_carry<<<dim3(1), dim3(BATCH * D_STATE), 0, stream>>>(Apow, carry_end, s_in);
    // K5: fix-up -> Bu now holds full states
    ssm_k5_fixup<<<dim3(BATCH * NCHUNK), dim3(D_STATE), 0, stream>>>(A, s_in, Bu);
    // K6: y = states @ C^T + GELU + LayerNorm
    ssm_k6_out<<<dim3(M_TOT / 16), dim3(256), 0, stream>>>(Bu, Cm, gamma, beta, out);
}